// TopkRouter_70257075028649
// MI455X (gfx1250) — compile-verified
//
#include <hip/hip_runtime.h>
#include <math.h>

// ---------------------------------------------------------------------------
// TopK expert router for MI455X (gfx1250, wave32).
//   scores = x @ W^T + b   (f32 WMMA 16x16x4),  per-token top-8 + masked softmax
// Bandwidth-bound: 268MB input stream @ 23.3TB/s ~ 12us floor; W (512KB) lives
// in L2. f32 WMMA keeps exact reference precision (no reason to downcast when
// compute is free under the HBM stream).
// ---------------------------------------------------------------------------

typedef float v2f __attribute__((ext_vector_type(2)));
typedef float v8f __attribute__((ext_vector_type(8)));

#define EMBED            2048
#define NEXP             64
#define ACTIVE           8
#define TOKENS_PER_WAVE  32          // 2 M-tiles of 16
#define WAVES_PER_BLOCK  4
#define BLOCK_THREADS    (WAVES_PER_BLOCK * 32)
#define LDS_STRIDE       65          // 64 + 1 pad -> conflict-free column reads

__global__ __launch_bounds__(BLOCK_THREADS)
void topk_router_wmma_kernel(const float* __restrict__ x,
                             const float* __restrict__ Wm,
                             const float* __restrict__ bias,
                             float* __restrict__ router_out,
                             float* __restrict__ indices_out)
{
    __shared__ float smem[WAVES_PER_BLOCK * TOKENS_PER_WAVE * LDS_STRIDE];

    const int lane = threadIdx.x & 31;
    const int wave = threadIdx.x >> 5;
    const int l16  = lane & 15;
    const int half = lane >> 4;                 // 0: lanes 0-15, 1: lanes 16-31

    const size_t tokenBase =
        ((size_t)blockIdx.x * WAVES_PER_BLOCK + wave) * TOKENS_PER_WAVE;

    // --- A fragments: 16x4 f32 tile. lane<16: K=k0,k0+1 in v0/v1 ; lane>=16: K=k0+2,k0+3
    const float* aBase0 = x + (tokenBase +      l16) * (size_t)EMBED + 2 * half;
    const float* aBase1 = x + (tokenBase + 16 + l16) * (size_t)EMBED + 2 * half;
    // --- B fragments: 4x16 f32 tile of W^T; B[k][n] = W[n][k], mirrored layout
    const float* wBase0 = Wm + (size_t)(l16     ) * EMBED + 2 * half;
    const float* wBase1 = Wm + (size_t)(l16 + 16) * EMBED + 2 * half;
    const float* wBase2 = Wm + (size_t)(l16 + 32) * EMBED + 2 * half;
    const float* wBase3 = Wm + (size_t)(l16 + 48) * EMBED + 2 * half;

    v8f acc[2][4];
#pragma unroll
    for (int mt = 0; mt < 2; ++mt)
#pragma unroll
        for (int nt = 0; nt < 4; ++nt)
            acc[mt][nt] = (v8f){0.f, 0.f, 0.f, 0.f, 0.f, 0.f, 0.f, 0.f};

    // ---- GEMM main loop: K = 2048 in chunks of 4, 8 WMMAs / chunk ----
    for (int k0 = 0; k0 < EMBED; k0 += 64) {
        // run-ahead prefetch on the streaming (A) operand -> global_prefetch_b8
        __builtin_prefetch(aBase0 + k0 + 256, 0, 0);
        __builtin_prefetch(aBase1 + k0 + 256, 0, 0);
#pragma unroll
        for (int kk = 0; kk < 64; kk += 4) {
            const int k = k0 + kk;
            v2f A0 = *(const v2f*)(aBase0 + k);
            v2f A1 = *(const v2f*)(aBase1 + k);
            v2f B0 = *(const v2f*)(wBase0 + k);
            v2f B1 = *(const v2f*)(wBase1 + k);
            v2f B2 = *(const v2f*)(wBase2 + k);
            v2f B3 = *(const v2f*)(wBase3 + k);

            acc[0][0] = __builtin_amdgcn_wmma_f32_16x16x4_f32(false, A0, false, B0, (short)0, acc[0][0], false, false);
            acc[0][1] = __builtin_amdgcn_wmma_f32_16x16x4_f32(false, A0, false, B1, (short)0, acc[0][1], false, false);
            acc[0][2] = __builtin_amdgcn_wmma_f32_16x16x4_f32(false, A0, false, B2, (short)0, acc[0][2], false, false);
            acc[0][3] = __builtin_amdgcn_wmma_f32_16x16x4_f32(false, A0, false, B3, (short)0, acc[0][3], false, false);
            acc[1][0] = __builtin_amdgcn_wmma_f32_16x16x4_f32(false, A1, false, B0, (short)0, acc[1][0], false, false);
            acc[1][1] = __builtin_amdgcn_wmma_f32_16x16x4_f32(false, A1, false, B1, (short)0, acc[1][1], false, false);
            acc[1][2] = __builtin_amdgcn_wmma_f32_16x16x4_f32(false, A1, false, B2, (short)0, acc[1][2], false, false);
            acc[1][3] = __builtin_amdgcn_wmma_f32_16x16x4_f32(false, A1, false, B3, (short)0, acc[1][3], false, false);
        }
    }

    // ---- bias add + spill scores to LDS ----
    // C/D layout: n = lane&15 (all VGPRs), m = r + 8*(lane>>4)
    float* waveLds = smem + wave * (TOKENS_PER_WAVE * LDS_STRIDE);
#pragma unroll
    for (int nt = 0; nt < 4; ++nt) {
        const float bn = bias[nt * 16 + l16];
#pragma unroll
        for (int mt = 0; mt < 2; ++mt) {
#pragma unroll
            for (int r = 0; r < 8; ++r) {
                const int tloc = mt * 16 + r + 8 * half;
                waveLds[tloc * LDS_STRIDE + nt * 16 + l16] = acc[mt][nt][r] + bn;
            }
        }
    }
    __syncthreads();

    // ---- per-token top-8 + masked softmax (one token per lane) ----
    const float* my = waveLds + lane * LDS_STRIDE;

    unsigned long long used = 0ull;
    float vmax = -INFINITY;
    for (int it = 0; it < ACTIVE; ++it) {
        float best = -INFINITY;
        int   bi   = 0;
        for (int e = 0; e < NEXP; ++e) {
            if (!((used >> e) & 1ull)) {
                const float v = my[e];
                if (v > best) { best = v; bi = e; }   // strict > => first-index tie-break (lax.top_k)
            }
        }
        used |= (1ull << bi);
        if (it == 0) vmax = best;                     // global max == max of selected set
    }

    float sum = 0.f;
    for (int e = 0; e < NEXP; ++e)
        if ((used >> e) & 1ull) sum += expf(my[e] - vmax);
    const float rsum = 1.f / sum;

    const size_t tok = tokenBase + (size_t)lane;
    float* ro = router_out  + tok * NEXP;
    float* io = indices_out + tok * NEXP;
#pragma unroll
    for (int e = 0; e < NEXP; e += 4) {
        float4 rv, iv;
        float* rvp = &rv.x;
        float* ivp = &iv.x;
#pragma unroll
        for (int j = 0; j < 4; ++j) {
            const int  ee  = e + j;
            const bool sel = (used >> ee) & 1ull;
            rvp[j] = sel ? expf(my[ee] - vmax) * rsum : 0.f;
            ivp[j] = sel ? 1.f : 0.f;
        }
        *(float4*)(ro + e) = rv;
        *(float4*)(io + e) = iv;
    }
}

extern "C" void kernel_launch(void* const* d_in, const int* in_sizes, int n_in,
                              void* d_out, int out_size, void* d_ws, size_t ws_size,
                              hipStream_t stream) {
    const float* x    = (const float*)d_in[0];   // [8,4096,2048]
    const float* Wm   = (const float*)d_in[1];   // [64,2048]
    const float* bias = (const float*)d_in[2];   // [64]

    const int total_tokens = in_sizes[0] / EMBED;          // 32768
    float* router  = (float*)d_out;                        // [T,64]
    float* indices = router + (size_t)total_tokens * NEXP; // [T,64]

    const int blocks = total_tokens / (TOKENS_PER_WAVE * WAVES_PER_BLOCK); // 256
    topk_router_wmma_kernel<<<blocks, BLOCK_THREADS, 0, stream>>>(
        x, Wm, bias, router, indices);
}